// GATModel_33818572488718
// MI455X (gfx1250) — compile-verified
//
#include <hip/hip_runtime.h>
#include <hip/hip_bf16.h>
#include <math.h>

#define NN   50000
#define EE   800000
#define E2T  (EE + NN)      // 850000 edges incl. self loops
#define FIN  128
#define HIDC 128
#define HEADS 2
#define HC1  (HEADS * HIDC) // 256
#define GG   1024

typedef float v2f __attribute__((ext_vector_type(2)));
typedef float v8f __attribute__((ext_vector_type(8)));

// ---------------- utility ----------------
__global__ void zero_f32(float* __restrict__ p, int n) {
    int i = blockIdx.x * blockDim.x + threadIdx.x;
    if (i < n) p[i] = 0.0f;
}

__device__ __forceinline__ unsigned fenc(float f) {
    unsigned u = __float_as_uint(f);
    return (u & 0x80000000u) ? ~u : (u | 0x80000000u);
}
__device__ __forceinline__ float fdec(unsigned e) {
    unsigned u = (e & 0x80000000u) ? (e ^ 0x80000000u) : ~e;
    return __uint_as_float(u);
}

// ---------------- fp32 WMMA GEMM: C[M,Nc] = A[M,K] @ B[K,Nc] ----------------
// One wave32 per 16x16 output tile. M, K, Nc all multiples of 16 / 4.
__global__ void gemm_wmma_f32(const float* __restrict__ A, const float* __restrict__ B,
                              float* __restrict__ C, int M, int K, int Nc) {
    int wid  = (blockIdx.x * blockDim.x + threadIdx.x) >> 5;
    int lane = threadIdx.x & 31;
    int tnc  = Nc >> 4;
    int tiles = (M >> 4) * tnc;
    if (wid >= tiles) return;                  // wave-uniform exit (EXEC stays full)
    int tm = wid / tnc, tn = wid % tnc;
    int half = lane >> 4, l = lane & 15;

    const float* arow = A + (size_t)(tm * 16 + l) * K;   // A: lane l -> row M=l
    const float* bcol = B + (size_t)(tn * 16 + l);       // B: lane l -> col N=l

    v8f acc = {};
    for (int k = 0; k < K; k += 4) {
        int ka = k + 2 * half;      // VGPR0: K=0 (lanes 0-15) / K=2 (lanes 16-31)
        v2f a, b;
        a.x = arow[ka];
        a.y = arow[ka + 1];         // VGPR1: K=1 / K=3
        b.x = bcol[(size_t)ka * Nc];
        b.y = bcol[(size_t)(ka + 1) * Nc];
        acc = __builtin_amdgcn_wmma_f32_16x16x4_f32(
            /*neg_a=*/false, a, /*neg_b=*/false, b,
            /*c_mod=*/(short)0, acc, /*reuse_a=*/false, /*reuse_b=*/false);
    }
    // D layout: VGPR r -> row tm*16 + r + 8*half, col tn*16 + l
    float* crow = C + (size_t)(tm * 16 + 8 * half) * Nc + tn * 16 + l;
#pragma unroll
    for (int r = 0; r < 8; ++r) crow[(size_t)r * Nc] = acc[r];
}

// ---------------- per-node attention coefficients ----------------
__global__ void node_alpha(const float* __restrict__ h, const float* __restrict__ a_src,
                           const float* __restrict__ a_dst, float* __restrict__ as,
                           float* __restrict__ ad, int n, int H, int C) {
    int i = blockIdx.x * blockDim.x + threadIdx.x;   // node*H + head
    if (i >= n * H) return;
    int hd = i % H;
    const float* hp = h + (size_t)i * C;
    float s = 0.f, d = 0.f;
    for (int c = 0; c < C; ++c) {
        float v = hp[c];
        s += v * a_src[hd * C + c];
        d += v * a_dst[hd * C + c];
    }
    as[i] = s;
    ad[i] = d;
}

// ---------------- edge logits + segment max (encoded atomicMax) ----------------
__global__ void edge_logits(const long long* __restrict__ ei, const float* __restrict__ as,
                            const float* __restrict__ ad, float* __restrict__ e_out,
                            unsigned* __restrict__ emax_enc, int E_, int Etot, int H) {
    int t = blockIdx.x * blockDim.x + threadIdx.x;
    if (t >= Etot * H) return;
    int e = t / H, hd = t - e * H;
    int s, d;
    if (e < E_) { s = (int)ei[e]; d = (int)ei[E_ + e]; }
    else        { s = d = e - E_; }
    float v = as[s * H + hd] + ad[d * H + hd];
    v = v > 0.f ? v : 0.2f * v;                  // leaky_relu, slope 0.2
    e_out[t] = v;
    atomicMax(&emax_enc[d * H + hd], fenc(v));
}

__global__ void decode_max(float* __restrict__ emax, int n) {
    int i = blockIdx.x * blockDim.x + threadIdx.x;
    if (i < n) emax[i] = fdec(__float_as_uint(emax[i]));
}

// ---------------- exp(e - max) + segment-sum denominator ----------------
__global__ void edge_exp(const long long* __restrict__ ei, const float* __restrict__ emax,
                         float* __restrict__ e_inout, float* __restrict__ den,
                         int E_, int Etot, int H) {
    int t = blockIdx.x * blockDim.x + threadIdx.x;
    if (t >= Etot * H) return;
    int e = t / H, hd = t - e * H;
    int d = (e < E_) ? (int)ei[E_ + e] : (e - E_);
    float ex = expf(e_inout[t] - emax[d * H + hd]);
    e_inout[t] = ex;
    atomicAdd(&den[d * H + hd], ex);
}

// ---------------- message aggregation: one block per edge ----------------
__global__ void edge_aggregate(const float* __restrict__ h, const float* __restrict__ ex,
                               const float* __restrict__ den, const long long* __restrict__ ei,
                               float* __restrict__ agg, int E_, int H, int C) {
    int e = blockIdx.x;
    int j = threadIdx.x;                 // 0 .. H*C-1
    int s, d;
    if (e < E_) { s = (int)ei[e]; d = (int)ei[E_ + e]; }
    else        { s = d = e - E_; }
    int hd = j / C;                      // wave-uniform (C is 128)
    float alpha = ex[(size_t)e * H + hd] / den[(size_t)d * H + hd];
    atomicAdd(&agg[(size_t)d * H * C + j], h[(size_t)s * H * C + j] * alpha);
}

// ---------------- bias + relu (in place) ----------------
__global__ void bias_relu(float* __restrict__ x, const float* __restrict__ b,
                          int total, int HC) {
    int i = blockIdx.x * blockDim.x + threadIdx.x;
    if (i >= total) return;
    float v = x[i] + b[i % HC];
    x[i] = v > 0.f ? v : 0.f;
}

// ---------------- global mean pool (sum + count) ----------------
__global__ void pool_kernel(const float* __restrict__ h, const long long* __restrict__ batch,
                            float* __restrict__ pooled, float* __restrict__ cnt, int n, int C) {
    int i = blockIdx.x * blockDim.x + threadIdx.x;
    if (i >= n * C) return;
    int node = i / C, c = i - node * C;
    int g = (int)batch[node];
    atomicAdd(&pooled[(size_t)g * C + c], h[i]);
    if (c == 0) atomicAdd(&cnt[g], 1.0f);
}

// ---------------- MLP head ----------------
__global__ void mlp1_kernel(const float* __restrict__ pooled, const float* __restrict__ cnt,
                            const float* __restrict__ w, const float* __restrict__ b,
                            float* __restrict__ z) {
    int i = blockIdx.x * blockDim.x + threadIdx.x;
    if (i >= GG * 64) return;
    int g = i >> 6, j = i & 63;
    float inv = 1.0f / fmaxf(cnt[g], 1.0f);
    float s = b[j];
    for (int k = 0; k < HIDC; ++k) s += pooled[(size_t)g * HIDC + k] * inv * w[k * 64 + j];
    z[i] = s > 0.f ? s : 0.f;
}

__global__ void mlp2_kernel(const float* __restrict__ z, const float* __restrict__ w,
                            const float* __restrict__ b, float* __restrict__ out) {
    int g = blockIdx.x * blockDim.x + threadIdx.x;
    if (g >= GG) return;
    float s = b[0];
    for (int j = 0; j < 64; ++j) s += z[(size_t)g * 64 + j] * w[j];
    out[g] = s;
}

// =======================================================================
extern "C" void kernel_launch(void* const* d_in, const int* in_sizes, int n_in,
                              void* d_out, int out_size, void* d_ws, size_t ws_size,
                              hipStream_t stream) {
    (void)in_sizes; (void)n_in; (void)out_size; (void)ws_size;

    const float*     x        = (const float*)d_in[0];
    const long long* ei       = (const long long*)d_in[1];   // [2,E] int64
    const long long* batch    = (const long long*)d_in[2];   // [N] int64
    const float*     W1       = (const float*)d_in[4];       // [128,256]
    const float*     att_src1 = (const float*)d_in[5];       // [2,128]
    const float*     att_dst1 = (const float*)d_in[6];
    const float*     b1       = (const float*)d_in[7];       // [256]
    const float*     W2       = (const float*)d_in[8];       // [256,128]
    const float*     att_src2 = (const float*)d_in[9];       // [1,128]
    const float*     att_dst2 = (const float*)d_in[10];
    const float*     b2       = (const float*)d_in[11];      // [128]
    const float*     lw1      = (const float*)d_in[12];      // [128,64]
    const float*     lb1      = (const float*)d_in[13];      // [64]
    const float*     lw2      = (const float*)d_in[14];      // [64,1]
    const float*     lb2      = (const float*)d_in[15];      // [1]
    float*           out      = (float*)d_out;               // [1024]

    // ---- workspace layout (floats) ----
    float* ws    = (float*)d_ws;
    float* h1    = ws;                                  // N*256
    float* agg1  = h1    + (size_t)NN * HC1;            // N*256
    float* h2    = agg1  + (size_t)NN * HC1;            // N*128
    float* as1   = h2    + (size_t)NN * HIDC;           // N*2
    float* ad1   = as1   + (size_t)NN * HEADS;          // N*2
    float* emax1 = ad1   + (size_t)NN * HEADS;          // N*2
    float* den1  = emax1 + (size_t)NN * HEADS;          // N*2
    float* e1    = den1  + (size_t)NN * HEADS;          // E2T*2
    float* as2   = e1    + (size_t)E2T * HEADS;         // N
    float* ad2   = as2   + NN;                          // N
    float* emax2 = ad2   + NN;                          // N
    float* den2  = emax2 + NN;                          // N
    float* e2    = den2  + NN;                          // E2T
    float* pooled= e2    + E2T;                         // G*128
    float* cnt   = pooled+ (size_t)GG * HIDC;           // G
    float* z1    = cnt   + GG;                          // G*64
    float* agg2  = h1;   // reuse h1 region after layer-1 aggregation is done

    auto cdiv = [](long long a, long long b) { return (int)((a + b - 1) / b); };
    const int BS = 256;

    // ================= Layer 1 =================
    {
        long long waves = (long long)(NN / 16) * (HC1 / 16);
        gemm_wmma_f32<<<cdiv(waves * 32, BS), BS, 0, stream>>>(x, W1, h1, NN, FIN, HC1);
    }
    node_alpha<<<cdiv((long long)NN * HEADS, BS), BS, 0, stream>>>(h1, att_src1, att_dst1,
                                                                   as1, ad1, NN, HEADS, HIDC);
    zero_f32<<<cdiv((long long)NN * HEADS * 2, BS), BS, 0, stream>>>(emax1, NN * HEADS * 2); // emax1+den1
    zero_f32<<<cdiv((long long)NN * HC1, BS), BS, 0, stream>>>(agg1, NN * HC1);

    edge_logits<<<cdiv((long long)E2T * HEADS, BS), BS, 0, stream>>>(ei, as1, ad1, e1,
                                                                     (unsigned*)emax1, EE, E2T, HEADS);
    decode_max<<<cdiv((long long)NN * HEADS, BS), BS, 0, stream>>>(emax1, NN * HEADS);
    edge_exp<<<cdiv((long long)E2T * HEADS, BS), BS, 0, stream>>>(ei, emax1, e1, den1, EE, E2T, HEADS);
    edge_aggregate<<<E2T, HC1, 0, stream>>>(h1, e1, den1, ei, agg1, EE, HEADS, HIDC);
    bias_relu<<<cdiv((long long)NN * HC1, BS), BS, 0, stream>>>(agg1, b1, NN * HC1, HC1);

    // ================= Layer 2 =================
    {
        long long waves = (long long)(NN / 16) * (HIDC / 16);
        gemm_wmma_f32<<<cdiv(waves * 32, BS), BS, 0, stream>>>(agg1, W2, h2, NN, HC1, HIDC);
    }
    node_alpha<<<cdiv((long long)NN, BS), BS, 0, stream>>>(h2, att_src2, att_dst2,
                                                           as2, ad2, NN, 1, HIDC);
    zero_f32<<<cdiv((long long)NN * 2, BS), BS, 0, stream>>>(emax2, NN * 2);  // emax2+den2
    zero_f32<<<cdiv((long long)NN * HIDC, BS), BS, 0, stream>>>(agg2, NN * HIDC);

    edge_logits<<<cdiv((long long)E2T, BS), BS, 0, stream>>>(ei, as2, ad2, e2,
                                                             (unsigned*)emax2, EE, E2T, 1);
    decode_max<<<cdiv((long long)NN, BS), BS, 0, stream>>>(emax2, NN);
    edge_exp<<<cdiv((long long)E2T, BS), BS, 0, stream>>>(ei, emax2, e2, den2, EE, E2T, 1);
    edge_aggregate<<<E2T, HIDC, 0, stream>>>(h2, e2, den2, ei, agg2, EE, 1, HIDC);
    bias_relu<<<cdiv((long long)NN * HIDC, BS), BS, 0, stream>>>(agg2, b2, NN * HIDC, HIDC);

    // ================= Pool + MLP head =================
    zero_f32<<<cdiv((long long)GG * HIDC + GG, BS), BS, 0, stream>>>(pooled, GG * HIDC + GG);
    pool_kernel<<<cdiv((long long)NN * HIDC, BS), BS, 0, stream>>>(agg2, batch, pooled, cnt, NN, HIDC);
    mlp1_kernel<<<cdiv((long long)GG * 64, BS), BS, 0, stream>>>(pooled, cnt, lw1, lb1, z1);
    mlp2_kernel<<<cdiv((long long)GG, BS), BS, 0, stream>>>(z1, lw2, lb2, out);
}